// ContrastiveLoss_78451872629246
// MI455X (gfx1250) — compile-verified
//
#include <hip/hip_runtime.h>
#include <hip/hip_bf16.h>

// Contrastive loss, algebraically reduced to per-class sufficient statistics,
// with the segmented reduction performed on the CDNA5 WMMA path in full FP32
// (V_WMMA_F32_16X16X4_F32: A 16x4 f32, B 4x16 f32, C/D 16x16 f32).
//
//   loss = 2 * sum_c (2*n_c*Q_c - 2*S_c^2)  -  (2*N*Q - 2*S^2)
// where S_c/Q_c/n_c are per-class sum / sum-of-squares / count.
//
// Per K=4 chunk of elements:
//   A[m][k] = (label_k == m) ? 1 : 0      (one-hot, exact in f32)
//   B[k][n] = w1(n)*p_k^2 + w0(n)*p_k + w2(n)   (n=0 -> p, n=1 -> p^2, n=2 -> 1)
// so the accumulator D[m][0..2] = (S_m, Q_m, n_m) partials per wave.

typedef __attribute__((ext_vector_type(2))) float v2f;
typedef __attribute__((ext_vector_type(8))) float v8f;

#define WAVES_TOTAL 64
#define CLASSES     16   // padded to the WMMA M dimension (real classes: 10)
#define STATS       3    // columns used: S, Q, count

// One K=4 WMMA step, unguarded (indices known in-bounds). i0 is 8B aligned.
static __device__ __forceinline__ v8f wmma_step(const float* __restrict__ pred,
                                                const int*   __restrict__ labels,
                                                int i0, int rowcol,
                                                float w0, float w1, float w2,
                                                v8f acc)
{
    const float2 p = *(const float2*)(pred + i0);    // global_load_b64
    const int2   l = *(const int2*)(labels + i0);    // global_load_b64

    // A 16x4 f32 layout (ISA 7.12.2): VGPR0 = K{0|2}, VGPR1 = K{1|3}
    v2f a;
    a.x = (l.x == rowcol) ? 1.0f : 0.0f;
    a.y = (l.y == rowcol) ? 1.0f : 0.0f;

    // B 4x16 f32: branchless column statistic via 2 FMAs per element.
    v2f b;
    b.x = __builtin_fmaf(p.x, __builtin_fmaf(p.x, w1, w0), w2);
    b.y = __builtin_fmaf(p.y, __builtin_fmaf(p.y, w1, w0), w2);

    // 8 args: (neg_a, A, neg_b, B, c_mod, C, reuse_a, reuse_b)
    return __builtin_amdgcn_wmma_f32_16x16x4_f32(
        false, a, false, b, (short)0, acc, false, false);
}

__global__ __launch_bounds__(256)
void ContrastiveLoss_partial_kernel(const float* __restrict__ pred,
                                    const int*   __restrict__ labels,
                                    float*       __restrict__ ws,
                                    int n, int per_wave)
{
    const int tid    = blockIdx.x * blockDim.x + threadIdx.x;
    const int lane   = threadIdx.x & 31;
    const int wave   = tid >> 5;                 // global wave id, 0..WAVES_TOTAL-1
    const int base0  = wave * per_wave;

    const int half   = lane >> 4;                // 0: lanes 0-15, 1: lanes 16-31
    const int rowcol = lane & 15;                // A row (M) / B column (N)
    const int k0     = half * 2;                 // this lane's K pair: {k0, k0+1}

    // Per-lane statistic selector weights (hoisted out of the loop).
    const float w0 = (rowcol == 0) ? 1.0f : 0.0f;   // column 0: sum p
    const float w1 = (rowcol == 1) ? 1.0f : 0.0f;   // column 1: sum p^2
    const float w2 = (rowcol == 2) ? 1.0f : 0.0f;   // column 2: count

    v8f acc0 = {}, acc1 = {}, acc2 = {}, acc3 = {};

    if (base0 + per_wave <= n) {
        // Fast path: whole slice in bounds (wave-uniform), no guards.
        // 4 independent accumulators -> back-to-back WMMAs with no RAW hazard.
        for (int off = 0; off < per_wave; off += 16) {
            const int i = base0 + off + k0;
            acc0 = wmma_step(pred, labels, i + 0,  rowcol, w0, w1, w2, acc0);
            acc1 = wmma_step(pred, labels, i + 4,  rowcol, w0, w1, w2, acc1);
            acc2 = wmma_step(pred, labels, i + 8,  rowcol, w0, w1, w2, acc2);
            acc3 = wmma_step(pred, labels, i + 12, rowcol, w0, w1, w2, acc3);
        }
    } else {
        // Tail path: branchless clamping; OOB validity folded into the one-hot
        // (invalid K column of A is all-zero => contributes nothing).
        for (int off = 0; off < per_wave; off += 4) {
            const int i0 = base0 + off + k0;
            const int i1 = i0 + 1;
            const bool v0 = i0 < n, v1 = i1 < n;
            const int j0 = v0 ? i0 : 0;          // cndmask, loads unconditional
            const int j1 = v1 ? i1 : 0;

            const int   l0 = labels[j0], l1 = labels[j1];
            const float p0 = pred[j0],   p1 = pred[j1];

            v2f a, b;
            a.x = (v0 && l0 == rowcol) ? 1.0f : 0.0f;
            a.y = (v1 && l1 == rowcol) ? 1.0f : 0.0f;
            b.x = __builtin_fmaf(p0, __builtin_fmaf(p0, w1, w0), w2);
            b.y = __builtin_fmaf(p1, __builtin_fmaf(p1, w1, w0), w2);

            acc0 = __builtin_amdgcn_wmma_f32_16x16x4_f32(
                false, a, false, b, (short)0, acc0, false, false);
        }
    }

    // Fold the 4 accumulators (elementwise, 24 v_add_f32).
    v8f acc;
    #pragma unroll
    for (int v = 0; v < 8; ++v)
        acc[v] = (acc0[v] + acc1[v]) + (acc2[v] + acc3[v]);

    // D element (M = v + 8*half, N = rowcol) lives in acc[v] (ISA 7.12.2).
    // Lanes with column < STATS export the per-class partials; fixed slot per
    // wave -> bitwise-deterministic final reduction (no float atomics).
    if (rowcol < STATS) {
        #pragma unroll
        for (int v = 0; v < 8; ++v) {
            const int m = v + 8 * half;
            ws[(wave * CLASSES + m) * STATS + rowcol] = acc[v];
        }
    }
}

__global__ void ContrastiveLoss_finalize_kernel(const float* __restrict__ ws,
                                                float* __restrict__ out)
{
    if (threadIdx.x != 0 || blockIdx.x != 0) return;

    float S[CLASSES], Q[CLASSES], C[CLASSES];
    #pragma unroll
    for (int c = 0; c < CLASSES; ++c) { S[c] = 0.0f; Q[c] = 0.0f; C[c] = 0.0f; }

    // Fixed accumulation order over waves -> deterministic result.
    for (int w = 0; w < WAVES_TOTAL; ++w) {
        for (int c = 0; c < CLASSES; ++c) {
            S[c] += ws[(w * CLASSES + c) * STATS + 0];
            Q[c] += ws[(w * CLASSES + c) * STATS + 1];
            C[c] += ws[(w * CLASSES + c) * STATS + 2];
        }
    }

    float St = 0.0f, Qt = 0.0f, Ct = 0.0f, within = 0.0f;
    for (int c = 0; c < CLASSES; ++c) {
        St += S[c]; Qt += Q[c]; Ct += C[c];
        within += 2.0f * C[c] * Q[c] - 2.0f * S[c] * S[c];
    }
    const float total = 2.0f * Ct * Qt - 2.0f * St * St;  // sum over all i,j (diag = 0)
    out[0] = 2.0f * within - total;
}

extern "C" void kernel_launch(void* const* d_in, const int* in_sizes, int n_in,
                              void* d_out, int out_size, void* d_ws, size_t ws_size,
                              hipStream_t stream)
{
    const float* pred   = (const float*)d_in[0];
    const int*   labels = (const int*)d_in[1];
    float*       out    = (float*)d_out;
    float*       ws     = (float*)d_ws;   // needs WAVES_TOTAL*16*3*4 = 12 KB

    const int n = in_sizes[0];
    int per_wave = (n + WAVES_TOTAL - 1) / WAVES_TOTAL;
    per_wave = (per_wave + 15) & ~15;     // multiple of 16 (4 WMMA steps / iter)

    // 8 blocks x 256 threads = 64 waves; each wave owns a contiguous slice.
    ContrastiveLoss_partial_kernel<<<8, 256, 0, stream>>>(pred, labels, ws, n, per_wave);
    ContrastiveLoss_finalize_kernel<<<1, 32, 0, stream>>>(ws, out);
}